// STDPLinear_17583596110463
// MI455X (gfx1250) — compile-verified
//
#include <hip/hip_runtime.h>
#include <float.h>

// ---------------- problem constants ----------------
static constexpr int Bn  = 128;
static constexpr int INN = 8192;
static constexpr int OUTN = 4096;
static constexpr float DECAY  = 0.2f;
static constexpr float DTRACE = 0.99f;   // both trace decays
static constexpr float INHC   = 1.625f;
static constexpr float A1C    = 0.96f;
static constexpr float A2C    = 0.53f;
static constexpr float PLUSC  = 0.002f;
static constexpr float THRCAP = 350.0f;

typedef __attribute__((ext_vector_type(2))) float v2f;
typedef __attribute__((ext_vector_type(8))) float v8f;

// ======================================================================
// Kernel 1: i = x @ W^T   via V_WMMA_F32_16X16X4_F32
//   A[m=b][k] = x[b,k]  (row-major, K fast)   B[k][n=o] = W[o,k]
// Lane L = (r = L&15, h = L>>4):
//   A: v0 = A[r][kb+2h], v1 = A[r][kb+2h+1]   (float2 load)
//   B: v0 = B[kb+2h][r], v1 = B[kb+2h+1][r]   (float2 load from W row r)
// W is streamed once (128 MB) -> non-temporal loads so x / i_cur / trace
// stay resident in the 192 MB L2 for the later kernels.
// ======================================================================
__global__ __launch_bounds__(32)
void gemm_xwT(const float* __restrict__ X, const float* __restrict__ W,
              float* __restrict__ I) {
  const int nb = blockIdx.x * 16;          // OUT tile
  const int mb = blockIdx.y * 16;          // batch tile
  const int lane = threadIdx.x;
  const int r = lane & 15, h = lane >> 4;

  const float* ap = X + (mb + r) * INN + 2 * h;
  const float* bp = W + (nb + r) * INN + 2 * h;

  v8f acc = {};
  for (int kb = 0; kb < INN; kb += 16) {
#pragma unroll
    for (int u = 0; u < 4; ++u) {
      v2f a = *(const v2f*)(ap + kb + 4 * u);
      v2f b = __builtin_nontemporal_load((const v2f*)(bp + kb + 4 * u));
      acc = __builtin_amdgcn_wmma_f32_16x16x4_f32(
          false, a, false, b, (short)0, acc, false, false);
    }
  }
  // C/D layout: VGPR rr -> lanes0-15: M=rr,N=lane ; lanes16-31: M=rr+8
#pragma unroll
  for (int rr = 0; rr < 8; ++rr) {
    const int m = mb + rr + 8 * h;
    const int n = nb + r;
    I[m * OUTN + n] = acc[rr];
  }
}

// ======================================================================
// Kernel 2: LIF + hard reset + WTA(k=1, random tie-break) + inhibition.
// One block per batch row; 512 threads x 8 columns (coalesced o = j*512+t).
// ======================================================================
__global__ __launch_bounds__(512)
void lif_wta(const float* __restrict__ I, const float* __restrict__ Mem,
             const float* __restrict__ Thr, const float* __restrict__ Rnd,
             float* __restrict__ spikes, float* __restrict__ mem_out,
             float* __restrict__ thr_acc) {
  __shared__ float sP[512], sI[512], sS[512];
  const int b = blockIdx.x;
  const int t = threadIdx.x;

  float iv[8], mr[8], pv[8], sv[8];
  float pmax = 0.0f, imax = -FLT_MAX, smax = 0.0f;

#pragma unroll
  for (int j = 0; j < 8; ++j) {
    const int o = j * 512 + t;
    const float i_ = I[b * OUTN + o];
    const float m_ = Mem[b * OUTN + o] * DECAY + i_;
    const float s_ = (m_ >= Thr[o]) ? 1.0f : 0.0f;
    const float p_ = s_ * Rnd[b * OUTN + o];
    iv[j] = i_;
    mr[j] = (s_ > 0.0f) ? 0.0f : m_;     // hard reset
    pv[j] = p_;
    sv[j] = s_;
    pmax = fmaxf(pmax, p_);
    imax = fmaxf(imax, i_);
    smax = fmaxf(smax, s_);
  }
  sP[t] = pmax; sI[t] = imax; sS[t] = smax;
  __syncthreads();
  for (int off = 256; off > 0; off >>= 1) {
    if (t < off) {
      sP[t] = fmaxf(sP[t], sP[t + off]);
      sI[t] = fmaxf(sI[t], sI[t + off]);
      sS[t] = fmaxf(sS[t], sS[t + off]);
    }
    __syncthreads();
  }
  pmax = sP[0]; imax = sI[0]; smax = sS[0];

  const float inh = INHC * imax * smax;   // applied only when the row spiked
#pragma unroll
  for (int j = 0; j < 8; ++j) {
    const int o = j * 512 + t;
    const float win = sv[j] * ((pv[j] >= pmax) ? 1.0f : 0.0f);
    spikes[b * OUTN + o]  = win;
    mem_out[b * OUTN + o] = mr[j] - inh * (1.0f - win);
    if (win > 0.0f) atomicAdd(&thr_acc[o], iv[j]);   // (i*spikes).sum(0)
  }
}

// ======================================================================
// Kernel 3: thr = threshold + PLUS*acc ; thr -= max(thr.max()-CAP, 0)
// ======================================================================
__global__ __launch_bounds__(1024)
void thr_final(const float* __restrict__ Thr, const float* __restrict__ acc,
               float* __restrict__ thr_out) {
  __shared__ float sM[1024];
  const int t = threadIdx.x;
  float v[4];
  float lmax = -FLT_MAX;
#pragma unroll
  for (int j = 0; j < 4; ++j) {
    const int o = j * 1024 + t;
    v[j] = Thr[o] + PLUSC * acc[o];
    lmax = fmaxf(lmax, v[j]);
  }
  sM[t] = lmax;
  __syncthreads();
  for (int off = 512; off > 0; off >>= 1) {
    if (t < off) sM[t] = fmaxf(sM[t], sM[t + off]);
    __syncthreads();
  }
  const float over = fmaxf(sM[0] - THRCAP, 0.0f);
#pragma unroll
  for (int j = 0; j < 4; ++j) thr_out[j * 1024 + t] = v[j] - over;
}

// ======================================================================
// Kernel 4: dw_total[o,i] = (A1*sum_b spikes[b,o]*(0.99*trace+x)[b,i]
//                          - A2*sum_b (0.99*trace2[b,o])*x[b,i]) / (tw*B)
// Four f32-WMMA accumulator chains (2 terms x 2 N-tiles) over K=B=128.
// The A operands (spikes^T, 0.99*trace2^T tiles) are loaded once per K-step
// and reused for both N-tiles: 12 loads per 4 WMMA. All operands read from
// the natural [B,OUT]/[B,IN] row-major arrays (64B-coalesced lane groups);
// trace decays fused into the loads. 134 MB output -> non-temporal stores.
// ======================================================================
__global__ __launch_bounds__(32)
void dw_gemm(const float* __restrict__ spk, const float* __restrict__ tr2,
             const float* __restrict__ tr,  const float* __restrict__ X,
             const int* __restrict__ twp, float* __restrict__ dw) {
  const int ib = blockIdx.x * 32;          // IN tiles (N, 2x16)
  const int ob = blockIdx.y * 16;          // OUT tile (M)
  const int lane = threadIdx.x;
  const int r = lane & 15, h = lane >> 4;

  v8f acc1a = {}, acc2a = {}, acc1b = {}, acc2b = {};
  for (int kb = 0; kb < Bn; kb += 4) {
    const int k0 = kb + 2 * h;
    // A1 = spikes^T tile, A2 = 0.99*trace2^T tile (M across lanes: coalesced)
    const float* s0 = spk + k0 * OUTN + ob + r;
    const float* t0 = tr2 + k0 * OUTN + ob + r;
    v2f a1, a2;
    a1.x = s0[0];            a1.y = s0[OUTN];
    a2.x = t0[0] * DTRACE;   a2.y = t0[OUTN] * DTRACE;
    // B1 = trace_new tiles, B2 = x tiles (N across lanes: coalesced)
    const float* r0 = tr + k0 * INN + ib + r;
    const float* x0 = X  + k0 * INN + ib + r;
    v2f b1a, b2a, b1b, b2b;
    b2a.x = x0[0];                          b2a.y = x0[INN];
    b2b.x = x0[16];                         b2b.y = x0[INN + 16];
    b1a.x = fmaf(r0[0],        DTRACE, b2a.x);
    b1a.y = fmaf(r0[INN],      DTRACE, b2a.y);
    b1b.x = fmaf(r0[16],       DTRACE, b2b.x);
    b1b.y = fmaf(r0[INN + 16], DTRACE, b2b.y);

    acc1a = __builtin_amdgcn_wmma_f32_16x16x4_f32(
        false, a1, false, b1a, (short)0, acc1a, false, false);
    acc2a = __builtin_amdgcn_wmma_f32_16x16x4_f32(
        false, a2, false, b2a, (short)0, acc2a, false, false);
    acc1b = __builtin_amdgcn_wmma_f32_16x16x4_f32(
        false, a1, false, b1b, (short)0, acc1b, false, false);
    acc2b = __builtin_amdgcn_wmma_f32_16x16x4_f32(
        false, a2, false, b2b, (short)0, acc2b, false, false);
  }

  const int tw = *twp;
  const float inv = 1.0f / (float)(tw * Bn);
  const float c1 = A1C * inv, c2 = A2C * inv;
#pragma unroll
  for (int rr = 0; rr < 8; ++rr) {
    const int o = ob + rr + 8 * h;
    const int i = ib + r;
    __builtin_nontemporal_store(c1 * acc1a[rr] - c2 * acc2a[rr],
                                dw + o * INN + i);
    __builtin_nontemporal_store(c1 * acc1b[rr] - c2 * acc2b[rr],
                                dw + o * INN + i + 16);
  }
}

// ---------------- small helper: zero the threshold accumulator ----------------
__global__ __launch_bounds__(256)
void zero_acc(float* __restrict__ p) {
  p[blockIdx.x * 256 + threadIdx.x] = 0.0f;
}

// ======================================================================
extern "C" void kernel_launch(void* const* d_in, const int* in_sizes, int n_in,
                              void* d_out, int out_size, void* d_ws, size_t ws_size,
                              hipStream_t stream) {
  const float* x      = (const float*)d_in[0];   // [128,8192]
  const float* w      = (const float*)d_in[1];   // [4096,8192]
  const float* mem    = (const float*)d_in[2];   // [128,4096]
  const float* thr    = (const float*)d_in[3];   // [4096]
  const float* trace  = (const float*)d_in[4];   // [128,8192]
  const float* trace2 = (const float*)d_in[5];   // [128,4096]
  const float* rnd    = (const float*)d_in[6];   // [128,4096]
  const int*   twp    = (const int*)d_in[7];     // scalar

  float* out    = (float*)d_out;
  float* spikes = out;                                   // 524288
  float* dwt    = out + 524288;                          // 33554432
  float* thro   = out + 524288 + 33554432;               // 4096
  float* memo   = out + 524288 + 33554432 + 4096;        // 524288

  float* i_cur   = (float*)d_ws;                         // [128,4096] f32
  float* thr_acc = i_cur + (size_t)Bn * OUTN;            // [4096] f32

  zero_acc<<<OUTN / 256, 256, 0, stream>>>(thr_acc);
  gemm_xwT<<<dim3(OUTN / 16, Bn / 16), 32, 0, stream>>>(x, w, i_cur);
  lif_wta<<<Bn, 512, 0, stream>>>(i_cur, mem, thr, rnd, spikes, memo, thr_acc);
  thr_final<<<1, 1024, 0, stream>>>(thr, thr_acc, thro);
  dw_gemm<<<dim3(INN / 32, OUTN / 16), 32, 0, stream>>>(spikes, trace2, trace, x,
                                                        twp, dwt);
}